// LaplacianPolynomial_84035330113976
// MI455X (gfx1250) — compile-verified
//
#include <hip/hip_runtime.h>
#include <hip/hip_bf16.h>

#define N_NODES 50000
#define N_EDGES 800000
#define IN_FEATS 512
#define OUT_FEATS 64
#define KPOLY 3
#define NF (N_NODES * OUT_FEATS)   // 3,200,000
#define NK4 (IN_FEATS / 4)         // 128 k-steps of K=4

typedef float v2f __attribute__((ext_vector_type(2)));
typedef float v4f __attribute__((ext_vector_type(4)));
typedef float v8f __attribute__((ext_vector_type(8)));

// ---------------------------------------------------------------- utilities
__global__ void zero_f32_kernel(float* __restrict__ p, int n) {
    int i = blockIdx.x * blockDim.x + threadIdx.x;
    if (i < n) p[i] = 0.0f;
}

// deg[v] += 1 for each edge (u->v)
__global__ void degree_kernel(const int* __restrict__ dst, float* __restrict__ deg) {
    int e = blockIdx.x * blockDim.x + threadIdx.x;
    if (e < N_EDGES) atomicAdd(&deg[dst[e]], 1.0f);
}

// norm[v] = clamp(deg,1)^-0.5 (in place on the deg buffer)
__global__ void norm_kernel(float* __restrict__ deg_norm) {
    int v = blockIdx.x * blockDim.x + threadIdx.x;
    if (v < N_NODES) {
        float d = deg_norm[v];
        d = d < 1.0f ? 1.0f : d;
        deg_norm[v] = rsqrtf(d);
    }
}

// ---------------------------------------------------------------- W pre-pack
// wp[((k0*4 + wave)*32 + lane)] = the v2f B-fragment lane `lane` of wave `wave`
// needs at k-step k0: {W[k+koff, n], W[k+koff+1, n]}, koff=(lane>>4)*2,
// n = wave*16 + (lane&15). Makes the GEMM's B load one coalesced b64.
__global__ void pack_w_kernel(const float* __restrict__ w, float* __restrict__ wp) {
    int t = blockIdx.x * blockDim.x + threadIdx.x;   // NK4*4*32 = 16384
    if (t < NK4 * 4 * 32) {
        int lane = t & 31;
        int wv   = (t >> 5) & 3;
        int k0   = t >> 7;
        int koff = (lane >> 4) << 1;
        int n    = wv * 16 + (lane & 15);
        int k    = k0 * 4 + koff;
        wp[2 * t + 0] = w[(size_t)(k + 0) * OUT_FEATS + n];
        wp[2 * t + 1] = w[(size_t)(k + 1) * OUT_FEATS + n];
    }
}

// ---------------------------------------------------------------- GEMM: feat = x @ W
// 50000x512 @ 512x64, fp32 WMMA 16x16x4. Block = 128 threads (4 waves).
// Block covers 32 rows x 64 cols; wave w -> cols [16w,16w+16), two M-tiles.
#define LDS_STRIDE 516   // 512 + 4 pad: bank stride 4 (conflict-free), 16B rows

__global__ __launch_bounds__(128) void gemm_wmma_kernel(const float* __restrict__ x,
                                                        const float* __restrict__ wp,
                                                        float* __restrict__ feat) {
    __shared__ float As[32 * LDS_STRIDE];   // 66,048 B (CDNA5: up to 320KB/WG)

    const int r0  = blockIdx.x * 32;
    const int tid = threadIdx.x;

    // Stage 32x512 A tile (4096 float4s, 32 per thread), NT loads: x is
    // single-use streaming data, keep it out of L2 (L2 holds fs/h/edges).
    #pragma unroll
    for (int j = 0; j < 32; ++j) {
        int idx4 = j * 128 + tid;
        int row  = idx4 >> 7;             // 128 float4 per row
        int c4   = idx4 & 127;
        int grow = r0 + row;
        if (grow >= N_NODES) grow = N_NODES - 1;   // tail block: clamp (stores guarded)
        const v4f v = __builtin_nontemporal_load(
            (const v4f*)(x + (size_t)grow * IN_FEATS + (size_t)c4 * 4));
        *(v4f*)(&As[row * LDS_STRIDE + c4 * 4]) = v;
    }
    __syncthreads();

    const int wave = tid >> 5;
    const int lane = tid & 31;
    const int m    = lane & 15;
    const int koff = (lane >> 4) << 1;    // lanes 0-15: K{0,1}; lanes 16-31: K{2,3}
    const int n    = wave * 16 + m;

    const float* a0p = &As[m * LDS_STRIDE + koff];
    const float* a1p = &As[(m + 16) * LDS_STRIDE + koff];
    const v2f*   bp  = (const v2f*)wp + (size_t)wave * 32 + lane;

    v8f c0 = {}, c1 = {};
    #pragma unroll 4
    for (int k0 = 0; k0 < NK4; ++k0) {
        v2f a0 = *(const v2f*)(a0p + k0 * 4);       // ds_load_b64 (2addr-paired)
        v2f a1 = *(const v2f*)(a1p + k0 * 4);       // ds_load_b64 (2addr-paired)
        v2f b  = bp[(size_t)k0 * 128];              // coalesced global_load_b64
        c0 = __builtin_amdgcn_wmma_f32_16x16x4_f32(false, a0, false, b,
                                                   (short)0, c0, false, false);
        c1 = __builtin_amdgcn_wmma_f32_16x16x4_f32(false, a1, false, b,
                                                   (short)0, c1, false, false);
    }

    // C/D layout: VGPR r -> M=r (lanes 0-15) or M=8+r (lanes 16-31), N = lane&15.
    // Block-uniform tail test: only the last block takes the guarded path, so
    // the hot path is 16 straight stores with immediate offsets (one clause).
    const int mbase = (lane < 16) ? 0 : 8;
    float* out = feat + (size_t)(r0 + mbase) * OUT_FEATS + n;
    if (r0 + 32 <= N_NODES) {
        #pragma unroll
        for (int r = 0; r < 8; ++r) {
            out[(size_t)r * OUT_FEATS]        = c0[r];
            out[(size_t)(r + 16) * OUT_FEATS] = c1[r];
        }
    } else {
        #pragma unroll
        for (int r = 0; r < 8; ++r) {
            int row0 = r0 + mbase + r;
            if (row0 < N_NODES)      out[(size_t)r * OUT_FEATS]        = c0[r];
            if (row0 + 16 < N_NODES) out[(size_t)(r + 16) * OUT_FEATS] = c1[r];
        }
    }
}

// ---------------------------------------------------------------- poly init
// result = p0*feat ; fs = feat*norm ; h = 0
__global__ void init_kernel(const float* __restrict__ feat, const float* __restrict__ norm,
                            const float* __restrict__ pol, float* __restrict__ result,
                            float* __restrict__ fs, float* __restrict__ h) {
    int i = blockIdx.x * blockDim.x + threadIdx.x;
    if (i < NF) {
        float f  = feat[i];
        float nv = norm[i >> 6];
        result[i] = pol[0] * f;
        fs[i]     = f * nv;
        h[i]      = 0.0f;
    }
}

// ---------------------------------------------------------------- SpMM scatter
// h[dst[e]] += fs[src[e]]  -- thread = (edge, 4-feat chunk); float4 gather + 4 atomics.
// fs and h are 12.8 MB each -> L2-resident; atomics execute in L2.
__global__ __launch_bounds__(256) void scatter_kernel(const int* __restrict__ src,
                                                      const int* __restrict__ dst,
                                                      const float* __restrict__ fs,
                                                      float* __restrict__ h) {
    int t = blockIdx.x * blockDim.x + threadIdx.x;   // 12.8M < 2^31
    if (t >= N_EDGES * 16) return;
    int e  = t >> 4;
    int f4 = (t & 15) << 2;
    int s = src[e];
    int d = dst[e];
    const v4f v = *(const v4f*)(fs + (size_t)s * OUT_FEATS + f4);
    float* hp = h + (size_t)d * OUT_FEATS + f4;
    atomicAdd(hp + 0, v.x);
    atomicAdd(hp + 1, v.y);
    atomicAdd(hp + 2, v.z);
    atomicAdd(hp + 3, v.w);
}

// ---------------------------------------------------------------- poly update
// feat = feat - h*norm ; result += p[i]*feat ; fs = feat*norm (next) ; h = 0
__global__ void update_kernel(float* __restrict__ feat, float* __restrict__ h,
                              const float* __restrict__ norm, const float* __restrict__ pol,
                              int pi, float* __restrict__ result, float* __restrict__ fs) {
    int i = blockIdx.x * blockDim.x + threadIdx.x;
    if (i < NF) {
        float nv = norm[i >> 6];
        float f  = feat[i] - h[i] * nv;
        feat[i]  = f;
        result[i] += pol[pi] * f;
        fs[i]    = f * nv;
        h[i]     = 0.0f;
    }
}

// ---------------------------------------------------------------- launch
extern "C" void kernel_launch(void* const* d_in, const int* in_sizes, int n_in,
                              void* d_out, int out_size, void* d_ws, size_t ws_size,
                              hipStream_t stream) {
    const float* x    = (const float*)d_in[0];   // [50000, 512]
    const int*   src  = (const int*)d_in[1];     // [800000]
    const int*   dst  = (const int*)d_in[2];     // [800000]
    const float* w    = (const float*)d_in[3];   // [512, 64]
    const float* pol  = (const float*)d_in[4];   // [4]
    float* result     = (float*)d_out;           // [50000, 64]

    float* ws   = (float*)d_ws;
    float* norm = ws;                  // N_NODES floats (deg, then norm in place)
    float* wpak = ws + 51200;          // 32768 floats: pre-packed W fragments
    float* feat = wpak + 32768;        // NF floats
    float* fs   = feat + NF;           // NF floats
    float* h    = fs + NF;             // NF floats
    // total ~39 MB of workspace

    // 1) degree + normalization
    zero_f32_kernel<<<(N_NODES + 255) / 256, 256, 0, stream>>>(norm, N_NODES);
    degree_kernel<<<(N_EDGES + 255) / 256, 256, 0, stream>>>(dst, norm);
    norm_kernel<<<(N_NODES + 255) / 256, 256, 0, stream>>>(norm);

    // 2) pack W into WMMA B-fragment order, then feat = x @ W (fp32 WMMA)
    pack_w_kernel<<<(NK4 * 4 * 32 + 255) / 256, 256, 0, stream>>>(w, wpak);
    gemm_wmma_kernel<<<(N_NODES + 31) / 32, 128, 0, stream>>>(x, wpak, feat);

    // 3) polynomial init
    init_kernel<<<(NF + 255) / 256, 256, 0, stream>>>(feat, norm, pol, result, fs, h);

    // 4) K rounds of normalized SpMM + accumulate
    const int scatter_blocks = (N_EDGES * 16 + 255) / 256;  // 50000
    for (int i = 1; i <= KPOLY; ++i) {
        scatter_kernel<<<scatter_blocks, 256, 0, stream>>>(src, dst, fs, h);
        update_kernel<<<(NF + 255) / 256, 256, 0, stream>>>(feat, h, norm, pol, i,
                                                            result, fs);
    }
}